// Quantization_62148176773135
// MI455X (gfx1250) — compile-verified
//
#include <hip/hip_runtime.h>
#include <hip/hip_bf16.h>

// ---------------------------------------------------------------------------
// VQ dequantize (AQLM-style): out[v*4+j] = codebooks[cb(v)][codes[v]][j] * scales[v>>4]
// Pure HBM-bound streaming gather: ~228 MB traffic, ~45 MFLOP -> no WMMA path,
// optimize data movement (LDS-resident LUT, async LDS fill, B128 NT streams).
// ---------------------------------------------------------------------------

typedef __attribute__((ext_vector_type(4))) float v4f;
typedef __attribute__((ext_vector_type(4))) int   v4i;

constexpr int kRows   = 4096;
constexpr int kCols   = 11008;
constexpr int kNVecs  = kRows * kCols / 4;   // 11,272,192 4-float vecs
constexpr int kHalf   = kNVecs / 2;          // codebook 0 / codebook 1 boundary
constexpr int kLutN   = 2 * 256;             // 512 float4 entries = 8 KB in LDS

// ---- gfx1250 async global -> LDS copy (ASYNCcnt-tracked), guarded ----------
// Probe (round 1 stderr) showed the builtin's param 0 is a non-const AS1
// pointer to an int 4-vector, so cast to exactly that.
__device__ __forceinline__ void lds_fill_b128(const v4f* __restrict__ g, v4f* l) {
#if __has_builtin(__builtin_amdgcn_global_load_async_to_lds_b128)
  void* gv = const_cast<void*>(static_cast<const void*>(g));
  void* lv = static_cast<void*>(l);
  __builtin_amdgcn_global_load_async_to_lds_b128(
      (__attribute__((address_space(1))) v4i*)gv,
      (__attribute__((address_space(3))) v4i*)lv,
      /*imm offset*/ 0, /*cpol*/ 0);
#else
  *l = *g;  // fallback: vmem load + ds_store
#endif
}

__device__ __forceinline__ void lds_fill_wait() {
#if __has_builtin(__builtin_amdgcn_global_load_async_to_lds_b128)
#if __has_builtin(__builtin_amdgcn_s_wait_asynccnt)
  __builtin_amdgcn_s_wait_asynccnt(0);
#else
  asm volatile("s_wait_asynccnt 0" ::: "memory");
#endif
#endif
}

__global__ __launch_bounds__(256) void vq_dequant_kernel(
    const int*   __restrict__ codes,      // [kNVecs] (flat view of [2, kNVecs/2])
    const float* __restrict__ codebooks,  // [2*256*4]
    const float* __restrict__ scales,     // [kNVecs/16]
    v4f*         __restrict__ out)        // [kNVecs]
{
  // --- stage both codebooks (8 KB) into LDS once per workgroup ---
  __shared__ v4f lut[kLutN];
  const v4f* cb = (const v4f*)codebooks;
  for (int i = threadIdx.x; i < kLutN; i += 256) {
    lds_fill_b128(cb + i, lut + i);
  }
  lds_fill_wait();
  __syncthreads();

  const int tid    = blockIdx.x * 256 + threadIdx.x;
  const int stride = gridDim.x * 256;          // total threads

  // Grid-stride sweeps; 4 lane-contiguous slices per sweep so every code
  // load (b32) and output store (b128) is perfectly coalesced per wave32.
  for (int base = tid; base < kNVecs; base += stride * 4) {
    const int nxt = base + stride * 4;
    if (nxt < kNVecs) {
      __builtin_prefetch(codes + nxt, 0, 1);   // global_prefetch_b8 (next sweep)
    }
#pragma unroll
    for (int j = 0; j < 4; ++j) {
      const int v = base + j * stride;
      if (v < kNVecs) {
        const int   code = __builtin_nontemporal_load(codes + v);   // read-once
        const float s    = scales[v >> 4];                          // 16 vecs/block
        const int   ofs  = (v >= kHalf) ? 256 : 0;                  // codebook id
        const v4f   e    = lut[ofs + code];                         // ds_load_b128
        const v4f   r    = e * s;
        __builtin_nontemporal_store(r, out + v);                    // b128 NT stream
      }
    }
  }
}

extern "C" void kernel_launch(void* const* d_in, const int* in_sizes, int n_in,
                              void* d_out, int out_size, void* d_ws, size_t ws_size,
                              hipStream_t stream) {
  const int*   codes     = (const int*)d_in[0];    // [2, kNVecs/2] int32
  const float* codebooks = (const float*)d_in[1];  // [2, 256, 4]  f32
  const float* scales    = (const float*)d_in[2];  // [kNVecs/16, 1] f32
  v4f*         out       = (v4f*)d_out;            // [4096, 11008] f32

  (void)in_sizes; (void)n_in; (void)out_size; (void)d_ws; (void)ws_size;

  // 2048 blocks x 8 wave32 each; ~5.4 grid-stride sweeps over 11.27M vecs.
  dim3 grid(2048), block(256);
  vq_dequant_kernel<<<grid, block, 0, stream>>>(codes, codebooks, scales, out);
}